// DownSampleLocal_43267500540731
// MI455X (gfx1250) — compile-verified
//
#include <hip/hip_runtime.h>
#include <math.h>

#define B_  2
#define C_  128
#define N_  4096
#define K_  32
#define M_  2048
#define ND_ 2048

typedef float v2f __attribute__((ext_vector_type(2)));
typedef float v8f __attribute__((ext_vector_type(8)));

#define SCORE_SCALE 33554432.0f   /* 2^25 fixed point for deterministic scatter-add */

// ---------------------------------------------------------------------------
// K1: squared norms per point + zero the integer score accumulator
// ---------------------------------------------------------------------------
__global__ __launch_bounds__(256) void k_sqnorm(const float* __restrict__ x,
                                                float* __restrict__ sq,
                                                int* __restrict__ score_i) {
  int i = blockIdx.x * 256 + threadIdx.x;          // i in [0, B*N)
  if (i >= B_ * N_) return;
  int b = i / N_, n = i % N_;
  const float* xp = x + (size_t)b * C_ * N_ + n;
  float s = 0.f;
#pragma unroll 8
  for (int c = 0; c < C_; ++c) { float v = xp[(size_t)c * N_]; s += v * v; }
  sq[i] = s;
  score_i[i] = 0;
}

// ---------------------------------------------------------------------------
// K2: Wqk[row][col] = sum_o Wk[o][row] * Wq[o][col]   (tiny 128x128x128)
// ---------------------------------------------------------------------------
__global__ __launch_bounds__(256) void k_wqk(const float* __restrict__ Wq,
                                             const float* __restrict__ Wk,
                                             float* __restrict__ Wqk) {
  int i = blockIdx.x * 256 + threadIdx.x;          // 16384 outputs
  if (i >= C_ * C_) return;
  int row = i / C_, col = i % C_;
  float s = 0.f;
#pragma unroll 8
  for (int o = 0; o < C_; ++o) s += Wk[o * C_ + row] * Wq[o * C_ + col];
  Wqk[i] = s;
}

// ---------------------------------------------------------------------------
// K3: tiled transpose x (B,C,N) -> xT (B,N,C); coalesced both sides
// ---------------------------------------------------------------------------
__global__ __launch_bounds__(256) void k_transpose(const float* __restrict__ x,
                                                   float* __restrict__ xT) {
  __shared__ float tile[32][33];
  const int b  = blockIdx.z;
  const int n0 = blockIdx.x * 32;
  const int c0 = blockIdx.y * 32;
  const int tx = threadIdx.x & 31, ty = threadIdx.x >> 5;   // 32x8
#pragma unroll
  for (int i = 0; i < 32; i += 8)
    tile[ty + i][tx] = x[(size_t)b * C_ * N_ + (size_t)(c0 + ty + i) * N_ + n0 + tx];
  __syncthreads();
#pragma unroll
  for (int i = 0; i < 32; i += 8)
    xT[(size_t)b * N_ * C_ + (size_t)(n0 + ty + i) * C_ + c0 + tx] = tile[tx][ty + i];
}

// ---------------------------------------------------------------------------
// K4: qkT[b][n][o] = sum_c W[o][c] x[b][c][n]  via V_WMMA_F32_16X16X4_F32.
// B-operand from x (C,N): perfectly coalesced 64B row pairs.
// Output stored transposed: each lane writes its 8 accumulators as 2x b128.
// ---------------------------------------------------------------------------
__global__ __launch_bounds__(32) void k_gemm_tout(const float* __restrict__ W,
                                                  const float* __restrict__ X,
                                                  float* __restrict__ OT) {
  const int b  = blockIdx.z;
  const int n0 = blockIdx.x * 16;
  const int o0 = blockIdx.y * 16;
  const int lane = threadIdx.x;
  const int half = lane >> 4, lm = lane & 15;

  const float* xb   = X + (size_t)b * C_ * N_;
  const float* Arow = W + (size_t)(o0 + lm) * C_ + 2 * half;

  v8f acc = {0.f, 0.f, 0.f, 0.f, 0.f, 0.f, 0.f, 0.f};
#pragma unroll
  for (int kk = 0; kk < C_; kk += 4) {
    v2f a;  a.x = Arow[kk];  a.y = Arow[kk + 1];
    int r0 = kk + 2 * half;
    v2f bb; bb.x = xb[(size_t)r0 * N_ + n0 + lm];
            bb.y = xb[(size_t)(r0 + 1) * N_ + n0 + lm];
    acc = __builtin_amdgcn_wmma_f32_16x16x4_f32(false, a, false, bb,
                                                (short)0, acc, false, false);
  }
  float* ob = OT + ((size_t)b * N_ + n0 + lm) * C_ + o0 + 8 * half;
  float4 lo = make_float4(acc[0], acc[1], acc[2], acc[3]);
  float4 hi = make_float4(acc[4], acc[5], acc[6], acc[7]);
  *(float4*)(ob)     = lo;
  *(float4*)(ob + 4) = hi;
}

// ---------------------------------------------------------------------------
// K5: O[b][o][n] = sum_c W[o][c] XT[b][n][c]  (B-operand row-major pairs).
// Used for y = Wv * aggT. Output (C,N) for the column gather.
// ---------------------------------------------------------------------------
__global__ __launch_bounds__(32) void k_gemm_bt(const float* __restrict__ W,
                                                const float* __restrict__ XT,
                                                float* __restrict__ O) {
  const int b  = blockIdx.z;
  const int n0 = blockIdx.x * 16;
  const int o0 = blockIdx.y * 16;
  const int lane = threadIdx.x;
  const int half = lane >> 4, lm = lane & 15;

  const float* Arow = W + (size_t)(o0 + lm) * C_ + 2 * half;
  const float* Brow = XT + ((size_t)b * N_ + n0 + lm) * C_ + 2 * half;

  v8f acc = {0.f, 0.f, 0.f, 0.f, 0.f, 0.f, 0.f, 0.f};
#pragma unroll
  for (int kk = 0; kk < C_; kk += 4) {
    v2f a;  a.x = Arow[kk];  a.y = Arow[kk + 1];
    v2f bb; bb.x = Brow[kk]; bb.y = Brow[kk + 1];
    acc = __builtin_amdgcn_wmma_f32_16x16x4_f32(false, a, false, bb,
                                                (short)0, acc, false, false);
  }
  float* ob = O + (size_t)b * C_ * N_ + (size_t)(o0 + 8 * half) * N_ + n0 + lm;
#pragma unroll
  for (int r = 0; r < 8; ++r) ob[(size_t)r * N_] = acc[r];
}

// ---------------------------------------------------------------------------
// K6: KNN. One WG (8 waves) owns 16 rows of the distance matrix.
// Phase 1: Gram tiles via fp32 WMMA -> dist block (16 x 4096) in 256 KB LDS.
// Phase 2: per-row top-32 extraction; each lane caches 4 sub-block minima
// (32 entries each) so a removal only rescans 32 LDS entries.
// ---------------------------------------------------------------------------
extern __shared__ float ldsd[];

__global__ __launch_bounds__(256) void k_knn(const float* __restrict__ x,
                                             const float* __restrict__ sq,
                                             int* __restrict__ idxnb) {
  const int b    = blockIdx.y;
  const int n0   = blockIdx.x * 16;
  const int wave = threadIdx.x >> 5;
  const int lane = threadIdx.x & 31;
  const int half = lane >> 4, lm = lane & 15;

  const float* xb  = x + (size_t)b * C_ * N_;
  const float* sqb = sq + (size_t)b * N_;

  // A fragments for the fixed 16-row block (hoisted: 32 steps x v2f)
  v2f afrag[32];
#pragma unroll
  for (int s = 0; s < 32; ++s) {
    int r0 = 4 * s + 2 * half;
    afrag[s].x = xb[(size_t)r0 * N_ + n0 + lm];
    afrag[s].y = xb[(size_t)(r0 + 1) * N_ + n0 + lm];
  }
  float sqr[8];
#pragma unroll
  for (int r = 0; r < 8; ++r) sqr[r] = sqb[n0 + r + 8 * half];

  for (int m0 = wave * 16; m0 < N_; m0 += 8 * 16) {
    float sqc = sqb[m0 + lm];
    v8f acc = {0.f, 0.f, 0.f, 0.f, 0.f, 0.f, 0.f, 0.f};
#pragma unroll
    for (int s = 0; s < 32; ++s) {
      int r0 = 4 * s + 2 * half;
      v2f bb; bb.x = xb[(size_t)r0 * N_ + m0 + lm];
              bb.y = xb[(size_t)(r0 + 1) * N_ + m0 + lm];
      acc = __builtin_amdgcn_wmma_f32_16x16x4_f32(false, afrag[s], false, bb,
                                                  (short)0, acc, false, false);
    }
#pragma unroll
    for (int r = 0; r < 8; ++r)
      ldsd[(size_t)(r + 8 * half) * N_ + m0 + lm] = sqr[r] + sqc - 2.0f * acc[r];
  }
  __syncthreads();

  // Phase 2: each wave extracts top-32 for rows 2*wave and 2*wave+1
  for (int rr = 0; rr < 2; ++rr) {
    const int row = wave * 2 + rr;
    float* drow = ldsd + (size_t)row * N_;
    int* idxrow = idxnb + ((size_t)b * N_ + n0 + row) * K_;
    const int seg0 = lane * 128;

    float bv4[4]; int bi4[4];
#pragma unroll
    for (int s = 0; s < 4; ++s) {
      float bv = 3.4e38f; int bi = seg0 + s * 32;
      for (int t = 0; t < 32; ++t) {
        float v = drow[seg0 + s * 32 + t];
        if (v < bv) { bv = v; bi = seg0 + s * 32 + t; }
      }
      bv4[s] = bv; bi4[s] = bi;
    }
    for (int k = 0; k < K_; ++k) {
      float v = bv4[0]; int ii = bi4[0];
#pragma unroll
      for (int s = 1; s < 4; ++s)
        if (bv4[s] < v) { v = bv4[s]; ii = bi4[s]; }   // strict <: lower idx wins ties
#pragma unroll
      for (int off = 16; off > 0; off >>= 1) {
        float ov = __shfl_xor(v, off, 32);
        int   oi = __shfl_xor(ii, off, 32);
        if (ov < v || (ov == v && oi < ii)) { v = ov; ii = oi; }
      }
      if (lane == 0) idxrow[k] = ii;
      if ((ii >> 7) == lane) {               // owning lane removes + rescans 32
        drow[ii] = 3.4e38f;
        int s = (ii >> 5) & 3;
        float bv = 3.4e38f; int bi = seg0 + s * 32;
        for (int t = 0; t < 32; ++t) {
          float vv = drow[seg0 + s * 32 + t];
          if (vv < bv) { bv = vv; bi = seg0 + s * 32 + t; }
        }
        bv4[s] = bv; bi4[s] = bi;
      }
    }
  }
}

// ---------------------------------------------------------------------------
// K7: attention on transposed layouts. One wave = one point; lane = neighbor.
// energy_k = qkT[n,:]·(xT[j_k,:]-xT[n,:])/sqrt(D) with float4 streaming loads;
// softmax/std across lanes; deterministic fixed-point scatter-add into score;
// channel-parallel aggregation writes aggT[n,:] contiguously.
// ---------------------------------------------------------------------------
__global__ __launch_bounds__(256) void k_attn(const float* __restrict__ xT,
                                              const float* __restrict__ qkT,
                                              const int* __restrict__ idxnb,
                                              int* __restrict__ score_i,
                                              float* __restrict__ stdv,
                                              float* __restrict__ aggT) {
  __shared__ float a_s[8][K_];
  __shared__ int   j_s[8][K_];

  const int wave = threadIdx.x >> 5;
  const int lane = threadIdx.x & 31;
  const size_t pbase = (size_t)blockIdx.x * 8;
  const size_t p = pbase + wave;
  const int b = (int)(p / N_), n = (int)(p % N_);

  const float* xTb = xT + (size_t)b * N_ * C_;
  const float* qn  = qkT + ((size_t)b * N_ + n) * C_;
  const float* xn  = xTb + (size_t)n * C_;
  const int j = idxnb[p * K_ + lane];
  const float* xj = xTb + (size_t)j * C_;

  float e = 0.f;
#pragma unroll 8
  for (int c = 0; c < C_; c += 4) {
    float4 q  = *(const float4*)(qn + c);
    float4 aj = *(const float4*)(xj + c);
    float4 an = *(const float4*)(xn + c);
    e += q.x * (aj.x - an.x) + q.y * (aj.y - an.y)
       + q.z * (aj.z - an.z) + q.w * (aj.w - an.w);
  }
  e *= 0.08838834764831845f;                  // 1/sqrt(128)

  float mx = e;
#pragma unroll
  for (int off = 16; off > 0; off >>= 1) mx = fmaxf(mx, __shfl_xor(mx, off, 32));
  float ex = __expf(e - mx);
  float sm = ex;
#pragma unroll
  for (int off = 16; off > 0; off >>= 1) sm += __shfl_xor(sm, off, 32);
  float a = ex / sm;

  // std over K (population)
  float tot = a;
#pragma unroll
  for (int off = 16; off > 0; off >>= 1) tot += __shfl_xor(tot, off, 32);
  float d = a - tot * (1.0f / 32.0f);
  float var = d * d;
#pragma unroll
  for (int off = 16; off > 0; off >>= 1) var += __shfl_xor(var, off, 32);
  if (lane == 0) stdv[p] = sqrtf(var * (1.0f / 32.0f));

  atomicAdd(&score_i[(size_t)b * N_ + j], __float2int_rn(a * SCORE_SCALE));

  a_s[wave][lane] = a;
  j_s[wave][lane] = j;
  __syncthreads();

  // channel-parallel aggregation: 2 points per pass, 4 passes, coalesced
  for (int t = 0; t < 4; ++t) {
    int pp = t * 2 + (threadIdx.x >> 7);
    int c  = threadIdx.x & 127;
    size_t pg = pbase + pp;
    int bb = (int)(pg / N_), nn = (int)(pg % N_);
    const float* xTb2 = xT + (size_t)bb * N_ * C_;
    float acc = -xTb2[(size_t)nn * C_ + c];
#pragma unroll
    for (int k = 0; k < K_; ++k) acc += a_s[pp][k] * xTb2[(size_t)j_s[pp][k] * C_ + c];
    aggT[((size_t)bb * N_ + nn) * C_ + c] = acc;
  }
}

// ---------------------------------------------------------------------------
// K8: exact top_k via full bitonic sort of 4096 (val,idx) pairs in LDS.
// which==0: score descending (ties: idx asc) -> first M = idx output
// which==1: std   ascending (ties: idx asc) -> first ND = idx_dropped
// ---------------------------------------------------------------------------
__global__ __launch_bounds__(1024) void k_sort(const int* __restrict__ score_i,
                                               const float* __restrict__ stdv,
                                               int* __restrict__ sel_ds,
                                               int* __restrict__ sel_dr,
                                               float* __restrict__ out_idx,
                                               float* __restrict__ out_idr) {
  __shared__ float sv[N_];
  __shared__ int   si[N_];
  const int b = blockIdx.x;
  const int which = blockIdx.y;
  const bool desc = (which == 0);

  for (int t = threadIdx.x; t < N_; t += 1024) {
    sv[t] = desc ? (float)score_i[(size_t)b * N_ + t] * (1.0f / SCORE_SCALE)
                 : stdv[(size_t)b * N_ + t];
    si[t] = t;
  }
  __syncthreads();

  for (int ksz = 2; ksz <= N_; ksz <<= 1) {
    for (int jg = ksz >> 1; jg > 0; jg >>= 1) {
      for (int t = threadIdx.x; t < N_ / 2; t += 1024) {
        int i   = ((t & ~(jg - 1)) << 1) | (t & (jg - 1));
        int ixj = i | jg;
        bool up = ((i & ksz) == 0);
        float av = sv[i], cv = sv[ixj];
        int   ai = si[i], ci = si[ixj];
        bool prec = desc ? (av > cv || (av == cv && ai < ci))
                         : (av < cv || (av == cv && ai < ci));
        bool doswap = up ? !prec : prec;
        if (doswap) { sv[i] = cv; si[i] = ci; sv[ixj] = av; si[ixj] = ai; }
      }
      __syncthreads();
    }
  }

  if (which == 0) {
    for (int t = threadIdx.x; t < M_; t += 1024) {
      sel_ds[(size_t)b * M_ + t] = si[t];
      out_idx[(size_t)b * M_ + t] = (float)si[t];
    }
  } else {
    for (int t = threadIdx.x; t < ND_; t += 1024) {
      sel_dr[(size_t)b * ND_ + t] = si[t];
      out_idr[(size_t)b * ND_ + t] = (float)si[t];
    }
  }
}

// ---------------------------------------------------------------------------
// K9: gather selected columns of y (C,N) into the output (B,C,m)
// ---------------------------------------------------------------------------
__global__ __launch_bounds__(256) void k_gather(const float* __restrict__ y,
                                                const int* __restrict__ sel,
                                                float* __restrict__ out,
                                                int mcount) {
  int i = blockIdx.x * 256 + threadIdx.x;          // [b][c][m]
  if (i >= B_ * C_ * mcount) return;
  int m = i % mcount; int t = i / mcount;
  int c = t % C_;     int b = t / C_;
  out[i] = y[(size_t)b * C_ * N_ + (size_t)c * N_ + sel[(size_t)b * mcount + m]];
}

// ---------------------------------------------------------------------------
extern "C" void kernel_launch(void* const* d_in, const int* in_sizes, int n_in,
                              void* d_out, int out_size, void* d_ws, size_t ws_size,
                              hipStream_t stream) {
  (void)in_sizes; (void)n_in; (void)out_size; (void)ws_size;
  const float* x  = (const float*)d_in[0];
  const float* Wq = (const float*)d_in[1];
  const float* Wk = (const float*)d_in[2];
  const float* Wv = (const float*)d_in[3];

  // workspace carve-up (~18 MB total)
  float* ws    = (float*)d_ws;
  float* sq    = ws;  ws += (size_t)B_ * N_;
  float* stdv  = ws;  ws += (size_t)B_ * N_;
  float* Wqk   = ws;  ws += (size_t)C_ * C_;
  float* xT    = ws;  ws += (size_t)B_ * N_ * C_;
  float* qkT   = ws;  ws += (size_t)B_ * N_ * C_;
  float* aggT  = ws;  ws += (size_t)B_ * N_ * C_;
  float* ybuf  = ws;  ws += (size_t)B_ * C_ * N_;
  int* score_i = (int*)ws;      ws += (size_t)B_ * N_;
  int* idxnb   = (int*)ws;      ws += (size_t)B_ * N_ * K_;
  int* sel_ds  = (int*)ws;      ws += (size_t)B_ * M_;
  int* sel_dr  = (int*)ws;      ws += (size_t)B_ * ND_;

  float* out_xds = (float*)d_out;                        // (B, C, M)
  float* out_idx = out_xds + (size_t)B_ * C_ * M_;       // (B, 1, M)
  float* out_xdr = out_idx + (size_t)B_ * M_;            // (B, C, ND)
  float* out_idr = out_xdr + (size_t)B_ * C_ * ND_;      // (B, 1, ND)

  k_sqnorm<<<(B_ * N_ + 255) / 256, 256, 0, stream>>>(x, sq, score_i);
  k_wqk<<<(C_ * C_ + 255) / 256, 256, 0, stream>>>(Wq, Wk, Wqk);
  k_transpose<<<dim3(N_ / 32, C_ / 32, B_), 256, 0, stream>>>(x, xT);
  k_gemm_tout<<<dim3(N_ / 16, C_ / 16, B_), 32, 0, stream>>>(Wqk, x, qkT);
  k_knn<<<dim3(N_ / 16, B_), 256, 16 * N_ * sizeof(float), stream>>>(x, sq, idxnb);
  k_attn<<<(B_ * N_) / 8, 256, 0, stream>>>(xT, qkT, idxnb, score_i, stdv, aggT);
  k_sort<<<dim3(B_, 2), 1024, 0, stream>>>(score_i, stdv, sel_ds, sel_dr, out_idx, out_idr);
  k_gemm_bt<<<dim3(N_ / 16, C_ / 16, B_), 32, 0, stream>>>(Wv, aggT, ybuf);
  k_gather<<<(B_ * C_ * M_ + 255) / 256, 256, 0, stream>>>(ybuf, sel_ds, out_xds, M_);
  k_gather<<<(B_ * C_ * ND_ + 255) / 256, 256, 0, stream>>>(ybuf, sel_dr, out_xdr, ND_);
}